// GateLayer_28200755265636
// MI455X (gfx1250) — compile-verified
//
#include <hip/hip_runtime.h>
#include <hip/hip_bf16.h>
#include <stdint.h>

#define M_DIM   4096    // row length of x (elements)
#define THREADS 256
#define RPB     8       // rows of x per block; LDS = RPB * 16KB = 128KB

typedef float vfloat4 __attribute__((ext_vector_type(4)));

// ---------------------------------------------------------------------------
// Kernel 1: per-gate metadata. softmax(alpha[g]) collapses to two coefficients:
//   y = (w0 - w1 - 2w2) * a*b + (w1 + w2) * (a + b)
// Pack {idx_l, idx_r, c_ab, c_s} into one uint4 for a single b128 load later.
// ---------------------------------------------------------------------------
__global__ __launch_bounds__(THREADS)
void gate_meta_kernel(const float* __restrict__ alpha,
                      const int*   __restrict__ idx_l,
                      const int*   __restrict__ idx_r,
                      uint4*       __restrict__ meta,
                      int G) {
    int g = blockIdx.x * blockDim.x + threadIdx.x;
    if (g >= G) return;
    float a0 = alpha[3 * g + 0];
    float a1 = alpha[3 * g + 1];
    float a2 = alpha[3 * g + 2];
    float m  = fmaxf(a0, fmaxf(a1, a2));
    float e0 = __expf(a0 - m);
    float e1 = __expf(a1 - m);
    float e2 = __expf(a2 - m);
    float inv = 1.0f / (e0 + e1 + e2);
    float w0 = e0 * inv, w1 = e1 * inv, w2 = e2 * inv;
    uint4 v;
    v.x = (unsigned)idx_l[g];
    v.y = (unsigned)idx_r[g];
    v.z = __float_as_uint(w0 - w1 - 2.0f * w2);   // c_ab
    v.w = __float_as_uint(w1 + w2);               // c_s
    meta[g] = v;
}

// ---------------------------------------------------------------------------
// Gate loop body, unswitched on whether all RPB rows are valid. NROWS is a
// compile-time row count for the hot path (RPB) and tail handling happens in
// the 1-row instantiation loop from the caller.
// ---------------------------------------------------------------------------
template <int NROWS>
__device__ __forceinline__
void gate_apply(const float* __restrict__ rows,   // LDS, NROWS*M_DIM floats
                const uint4* __restrict__ meta,
                float* __restrict__ out_base,     // y + row0*G
                int G, int tid) {
    // Hoist per-row output pointers out of the gate loop.
    float* out[NROWS];
#pragma unroll
    for (int r = 0; r < NROWS; ++r) out[r] = out_base + (size_t)r * (size_t)G;

    for (int g0 = tid * 4; g0 < G; g0 += THREADS * 4) {
        const uint4 m0 = meta[g0 + 0];
        const uint4 m1 = meta[g0 + 1];
        const uint4 m2 = meta[g0 + 2];
        const uint4 m3 = meta[g0 + 3];

        const float c0 = __uint_as_float(m0.z), d0 = __uint_as_float(m0.w);
        const float c1 = __uint_as_float(m1.z), d1 = __uint_as_float(m1.w);
        const float c2 = __uint_as_float(m2.z), d2 = __uint_as_float(m2.w);
        const float c3 = __uint_as_float(m3.z), d3 = __uint_as_float(m3.w);

#pragma unroll
        for (int r = 0; r < NROWS; ++r) {
            const float* lrow = &rows[r * M_DIM];
            vfloat4 o;
            {
                float a = lrow[m0.x], b = lrow[m0.y];
                o.x = fmaf(c0, a * b, d0 * (a + b));
            }
            {
                float a = lrow[m1.x], b = lrow[m1.y];
                o.y = fmaf(c1, a * b, d1 * (a + b));
            }
            {
                float a = lrow[m2.x], b = lrow[m2.y];
                o.z = fmaf(c2, a * b, d2 * (a + b));
            }
            {
                float a = lrow[m3.x], b = lrow[m3.y];
                o.w = fmaf(c3, a * b, d3 * (a + b));
            }
            __builtin_nontemporal_store(o, (vfloat4*)(out[r] + g0));
        }
    }
}

// ---------------------------------------------------------------------------
// Kernel 2: RPB batch rows per block. Stage RPB*16KB of x into LDS with CDNA5
// async global->LDS b128 loads (ASYNCcnt), then stream gate metadata ONCE per
// block and apply each gate to all RPB rows (8x metadata-bandwidth reduction).
// Output leaves through non-temporal b128 stores so the 256MB write stream
// does not evict x/meta from the 192MB L2.
// ---------------------------------------------------------------------------
__global__ __launch_bounds__(THREADS)
void gate_main_kernel(const float* __restrict__ x,
                      const uint4* __restrict__ meta,
                      float*       __restrict__ y,
                      int G, int B) {
    extern __shared__ float rows[];   // RPB * M_DIM floats = 128 KB

    const int tid  = threadIdx.x;
    const int row0 = blockIdx.x * RPB;
    const int nr   = (B - row0 < RPB) ? (B - row0) : RPB;   // uniform

    // Low 32 bits of the generic shared-pointer are the LDS byte address
    const unsigned lds_base = (unsigned)(uintptr_t)(&rows[0]);

    // Fill LDS: per row, 4096 floats = 16KB; 256 threads x 16B -> 4 passes.
    // All async issues outstanding before one wait -> max MLP on the fill.
#pragma unroll
    for (int r = 0; r < RPB; ++r) {
        if (r < nr) {
            const float* xrow = x + (size_t)(row0 + r) * M_DIM;
            const unsigned rbase = lds_base + (unsigned)(r * M_DIM * 4);
#pragma unroll
            for (int p = 0; p < 4; ++p) {
                unsigned off = (unsigned)(p * 4096 + tid * 16);
                asm volatile("global_load_async_to_lds_b128 %0, %1, %2"
                             :
                             : "v"(rbase + off), "v"(off), "s"(xrow)
                             : "memory");
            }
        }
    }
    asm volatile("s_wait_asynccnt 0x0" ::: "memory");
    __syncthreads();

    float* out_base = y + (size_t)row0 * (size_t)G;

    if (nr == RPB) {
        // Hot path: branch-free fully-unrolled 8-row body.
        gate_apply<RPB>(rows, meta, out_base, G, tid);
    } else {
        // Tail block: handle remaining rows one at a time.
        for (int r = 0; r < nr; ++r) {
            gate_apply<1>(&rows[r * M_DIM], meta,
                          out_base + (size_t)r * (size_t)G, G, tid);
        }
    }
}

// ---------------------------------------------------------------------------
// Host entry
// ---------------------------------------------------------------------------
extern "C" void kernel_launch(void* const* d_in, const int* in_sizes, int n_in,
                              void* d_out, int out_size, void* d_ws, size_t ws_size,
                              hipStream_t stream) {
    const float* x     = (const float*)d_in[0];   // (B, M) f32
    const int*   idx_l = (const int*)  d_in[1];   // (G,)
    const int*   idx_r = (const int*)  d_in[2];   // (G,)
    const float* alpha = (const float*)d_in[3];   // (G, 3) f32
    float*       y     = (float*)d_out;           // (B, G) f32
    uint4*       meta  = (uint4*)d_ws;            // G * 16 bytes

    const int G = in_sizes[1];
    const int B = in_sizes[0] / M_DIM;

    gate_meta_kernel<<<(G + THREADS - 1) / THREADS, THREADS, 0, stream>>>(
        alpha, idx_l, idx_r, meta, G);

    const int    nblocks   = (B + RPB - 1) / RPB;
    const size_t lds_bytes = (size_t)RPB * M_DIM * sizeof(float);   // 128 KB
    gate_main_kernel<<<nblocks, THREADS, lds_bytes, stream>>>(x, meta, y, G, B);
}